// CSBrain_43817256354442
// MI455X (gfx1250) — compile-verified
//
#include <hip/hip_runtime.h>

// MI455X / gfx1250, wave32.
// fp32 reference -> V_WMMA_F32_16X16X4_F32 (native fp32 matrix op, fp32 acc).
// Data path: async global->LDS DMA (GLOBAL_LOAD_ASYNC_TO_LDS_B128, ASYNCcnt)
// double-buffered against the WMMA K-loop; per-copy deltas live in the
// instruction's immediate offset (applies to both global and LDS sides).
//
// Mapping: per-(electrode c, batch b, t-half, n-tile) GEMM tile.
//   M = 64 t-rows (4 waves x 16), N = 16 channels,
//   K pipelined in chunks of 100 features; taps k in {1,3,5} accumulate.
// x (105 MB) is resident in the 192 MB L2, so per-tap re-reads are L2 hits.

typedef __attribute__((ext_vector_type(2))) float v2f;
typedef __attribute__((ext_vector_type(8))) float v8f;
typedef __attribute__((ext_vector_type(4))) int   v4i;

#define B_    16
#define C_    64
#define T_    128
#define F_    200   // DIM_IN
#define NOUT_ 200   // DIM_OUT

#define KC    100               // K chunk (half of F_)
#define LDS_A (64 * KC)         // one A buffer: 25600 B (row stride 100 floats)
#define LDS_W (100 * 32)        // 3200 floats; rows 0-49 = K-half 0, 50-99 = half 1

#define GLOBAL_AS __attribute__((address_space(1)))
#define LDS_AS    __attribute__((address_space(3)))

// 16B async DMA with compile-time instruction offset (bytes). ISA: IOFFSET is
// added to BOTH the global address and the LDS address, so one base pair +
// immediate deltas covers a whole strided copy sequence.
template <int OFF>
static __device__ __forceinline__ void async_copy16(const float* g, float* l) {
    __builtin_amdgcn_global_load_async_to_lds_b128(
        (GLOBAL_AS v4i*)(uintptr_t)g,
        (LDS_AS    v4i*)(uint32_t)(uintptr_t)l,
        OFF, 0);
}

__global__ __launch_bounds__(128) void region_conv_wmma(
    const float* __restrict__ x,
    const float* __restrict__ W1, const float* __restrict__ b1,
    const float* __restrict__ W3, const float* __restrict__ b3,
    const float* __restrict__ W5, const float* __restrict__ b5,
    float* __restrict__ out)
{
    __shared__ float sA[2][LDS_A];   // ping-pong K-chunk buffers
    __shared__ float sW[LDS_W];      // (f>>1)*32 + n*2 + (f&1) interleave

    const int tile = blockIdx.x;   // 0..14 : N-tile over the 3 sections
    const int by   = blockIdx.y;   // 0..31 : batch * 2 + t-half
    const int c    = blockIdx.z;   // 0..63 : electrode
    const int b    = by >> 1;
    const int t0   = (by & 1) * 64;

    // ---- section decode (uniform) ----
    int sec, n0, ksz, dsec, colbase;
    if (tile < 7)       { sec = 0; n0 = tile * 16;        ksz = 1; dsec = 100; colbase = 0;   }
    else if (tile < 11) { sec = 1; n0 = (tile - 7) * 16;  ksz = 3; dsec = 50;  colbase = 100; }
    else                { sec = 2; n0 = (tile - 11) * 16; ksz = 5; dsec = 50;  colbase = 150; }
    int nv = dsec - n0; if (nv > 16) nv = 16;   // valid cols in this tile
    const int pad = (ksz - 1) >> 1;

    const float* Wsec = (sec == 0) ? W1 : (sec == 1) ? W3 : W5;
    const float* bsec = (sec == 0) ? b1 : (sec == 1) ? b3 : b5;

    // ---- region lookup (uniform -> SALU) ----
    const int rstart[5] = {0, 12, 26, 38, 52};
    const int rsize[5]  = {12, 14, 12, 14, 12};
    int r = 0;
#pragma unroll
    for (int i = 1; i < 5; ++i) if (c >= rstart[i]) r = i;
    const int s0 = rstart[r];
    const int ne = rsize[r];
    const int e  = c - s0;

    // ---- lane decomposition ----
    const int tid   = threadIdx.x;   // 0..127
    const int wave  = tid >> 5;      // 0..3   -> M sub-tile
    const int lane  = tid & 31;
    const int lrow  = lane & 15;     // A row within 16 / B,C,D column N
    const int khalf = lane >> 4;     // A/B: K offset 0 or 2; C/D: row +8
    const int arow  = wave * 16 + lrow;

    const int Q = 2 * ksz;           // K chunks: (tap j, half h), q = 2j + h

    // stageA fixed per-thread geometry: row = tid>>1, parity picks odd/even
    // 16B column; 13 copies at byte offsets 0,32,...,384 (25 x b128 per row).
    const int a_row = tid >> 1;          // 0..63
    const int a_p   = tid & 1;           // 0 -> 13 copies, 1 -> 12 copies
    // stageW fixed per-thread geometry: nn = tid>>3 (16 cols), fl0 = tid&7;
    // per-l strides: global 8*ksz floats, LDS 128 floats (constant).
    const int w_nn  = tid >> 3;          // 0..15
    const int w_fl0 = tid & 7;           // 0..7
    const bool w_ok = w_nn < nv;

    // ---- async-stage A chunk q into sA[q&1]: 13 DMAs per wave, uniform ---
    auto stageA = [&](int q) {
        const int j = q >> 1, h = q & 1;
        const int el   = (e + j - pad + ne) % ne;     // circular within region
        const int csrc = s0 + el;
        const float* g = x + (((size_t)b * C_ + csrc) * T_ + t0) * F_
                           + (size_t)a_row * F_ + h * KC + a_p * 4;
        float* l = &sA[q & 1][a_row * KC + a_p * 4];
        async_copy16<0 * 32>(g, l);   async_copy16<1 * 32>(g, l);
        async_copy16<2 * 32>(g, l);   async_copy16<3 * 32>(g, l);
        async_copy16<4 * 32>(g, l);   async_copy16<5 * 32>(g, l);
        async_copy16<6 * 32>(g, l);   async_copy16<7 * 32>(g, l);
        async_copy16<8 * 32>(g, l);   async_copy16<9 * 32>(g, l);
        async_copy16<10 * 32>(g, l);  async_copy16<11 * 32>(g, l);
        if (a_p == 0)                 // 16 lanes/wave active: still issues
            async_copy16<12 * 32>(g, l);
    };

    // ---- stage W chunk q into its K-half of sW (zero-fill invalid cols) --
    auto stageW = [&](int q) {
        const int j = q >> 1, h = q & 1;
        const float* wg = Wsec +
            ((((size_t)r * dsec + (n0 + w_nn)) * F_ + h * KC + w_fl0) * ksz + j);
        float* wl = &sW[(h * 50 + (w_fl0 >> 1)) * 32 + w_nn * 2 + (w_fl0 & 1)];
        const int gstep = 8 * ksz;
#pragma unroll
        for (int l = 0; l < 13; ++l) {
            if (w_fl0 + 8 * l < KC) {                 // l==12: fl0<4 lanes only
                float wv = 0.0f;
                if (w_ok) wv = wg[(size_t)l * gstep]; // EXEC-masked, no OOB
                wl[l * 128] = wv;
            }
        }
    };

    v8f acc = {};

    stageA(0);
    stageW(0);
    for (int q = 0; q < Q; ++q) {
        // issue next chunk's DMA into the buffer vacated two chunks ago
        // (end-of-iteration barrier below makes that overwrite legal)
        if (q + 1 < Q) {
            stageA(q + 1);
            stageW(q + 1);
            // retire chunk q's 13 DMAs; chunk q+1's 13 stay in flight
            asm volatile("s_wait_asynccnt 0xd" ::: "memory");
        } else {
            asm volatile("s_wait_asynccnt 0x0" ::: "memory");
        }
        __syncthreads();   // all waves' chunk-q A (DMA) and W (ds) visible

        // ---- 25 x v_wmma_f32_16x16x4_f32 on chunk q (EXEC all-ones) ----
        const int h = q & 1;
        const float* ap = &sA[q & 1][arow * KC + khalf * 2];
        const float* bp = &sW[(h * 50 + khalf) * 32 + lrow * 2];
#pragma unroll
        for (int k0 = 0; k0 < KC; k0 += 4) {
            v2f a  = *(const v2f*)(ap + k0);             // A[m][kk], A[m][kk+1]
            v2f bb = *(const v2f*)(bp + (k0 >> 1) * 32); // B[kk][n], B[kk+1][n]
            acc = __builtin_amdgcn_wmma_f32_16x16x4_f32(
                false, a, false, bb, (short)0, acc, false, false);
        }
        __syncthreads();   // chunk-q buffers free for chunk q+2's DMA
    }

    // ---- bias + predicated store (C/D: vgpr i -> M=i+8*khalf, N=lrow) ----
    if (lrow < nv) {
        const float bias = bsec[(size_t)r * dsec + n0 + lrow];
        const int colg = colbase + n0 + lrow;
        float* obase = out + (((size_t)b * C_ + c) * T_) * NOUT_ + colg;
#pragma unroll
        for (int i = 0; i < 8; ++i) {
            const int m = t0 + wave * 16 + i + khalf * 8;
            obase[(size_t)m * NOUT_] = acc[i] + bias;
        }
    }
}

extern "C" void kernel_launch(void* const* d_in, const int* in_sizes, int n_in,
                              void* d_out, int out_size, void* d_ws, size_t ws_size,
                              hipStream_t stream) {
    const float* x  = (const float*)d_in[0];
    const float* W1 = (const float*)d_in[1];
    const float* b1 = (const float*)d_in[2];
    const float* W3 = (const float*)d_in[3];
    const float* b3 = (const float*)d_in[4];
    const float* W5 = (const float*)d_in[5];
    const float* b5 = (const float*)d_in[6];
    float* out = (float*)d_out;

    dim3 grid(15, 32, 64);   // n-tiles x (batch*2 t-halves) x electrodes
    dim3 block(128);         // 4 wave32
    hipLaunchKernelGGL(region_conv_wmma, grid, block, 0, stream,
                       x, W1, b1, W3, b3, W5, b5, out);
}